// GATModel_28089086116668
// MI455X (gfx1250) — compile-verified
//
#include <hip/hip_runtime.h>

typedef __attribute__((ext_vector_type(2))) float v2f;
typedef __attribute__((ext_vector_type(4))) float v4f;
typedef __attribute__((ext_vector_type(8))) float v8f;

#define NE 640000
#define C 128
#define NEG_SLOPE 0.2f

// ---------------------------------------------------------------------------
// Kernel 1: h = x @ W via V_WMMA_F32_16X16X4_F32, fused a_src/a_dst row dots.
// 256 threads = 8 waves; wave w computes rows [blk*128 + w*16, +16) x all 128 cols.
// LDS holds W pre-paired along K: ldsP[p*128 + col] = {W[2p][col], W[2p+1][col]}
// so each WMMA B fragment is ONE aligned ds_load_b64 (no re-packing moves).
// ---------------------------------------------------------------------------
__global__ __launch_bounds__(256) void gat_gemm_kernel(
    const float* __restrict__ x, const float* __restrict__ W,
    const float* __restrict__ att_src, const float* __restrict__ att_dst,
    float* __restrict__ h, float* __restrict__ a_src, float* __restrict__ a_dst)
{
    __shared__ v2f ldsP[(C / 2) * C];   // 64 KB: K-paired W tile

    const int tid = threadIdx.x;

    // Stage W -> paired LDS layout. 64 pair-rows x 32 col4-groups = 2048 chunks.
    {
        const v4f* W4 = (const v4f*)W;
        for (int idx = tid; idx < (C / 2) * (C / 4); idx += 256) {
            const int p  = idx >> 5;        // K-pair row (k = 2p, 2p+1)
            const int c4 = idx & 31;        // group of 4 columns
            v4f r0 = W4[(2 * p + 0) * (C / 4) + c4];
            v4f r1 = W4[(2 * p + 1) * (C / 4) + c4];
            v2f* dst = &ldsP[p * C + c4 * 4];
            v2f t0; t0.x = r0.x; t0.y = r1.x; dst[0] = t0;
            v2f t1; t1.x = r0.y; t1.y = r1.y; dst[1] = t1;
            v2f t2; t2.x = r0.z; t2.y = r1.z; dst[2] = t2;
            v2f t3; t3.x = r0.w; t3.y = r1.w; dst[3] = t3;
        }
    }
    __syncthreads();

    const int lane  = tid & 31;
    const int wave  = tid >> 5;
    const int m16   = lane & 15;   // position within 16-lane group
    const int khalf = lane >> 4;   // 0: holds K=k0,k0+1   1: holds K=k0+2,k0+3
    const int rowBase = blockIdx.x * 128 + wave * 16;

    // Per-lane slices of the attention vectors (column t*16 + m16).
    float attS[8], attD[8];
    #pragma unroll
    for (int t = 0; t < 8; ++t) {
        attS[t] = att_src[t * 16 + m16];
        attD[t] = att_dst[t * 16 + m16];
    }

    v8f acc[8] = {};   // 8 N-tiles of 16x16 f32 accumulators (64 VGPRs)

    // A-matrix 16x4 f32 layout: lanes 0-15 rows M, VGPR0/1 = K0,K1; lanes 16-31 K2,K3.
    const float* xrow = x + (size_t)(rowBase + m16) * C;

    for (int k0 = 0; k0 < C; k0 += 4) {
        v2f a = *(const v2f*)(xrow + k0 + khalf * 2);
        // B fragment pair-row: p = k0/2 + khalf ; element = {W[k][col], W[k+1][col]}
        const v2f* brow = &ldsP[((k0 >> 1) + khalf) * C + m16];
        #pragma unroll
        for (int t = 0; t < 8; ++t) {
            v2f b = brow[t * 16];
            acc[t] = __builtin_amdgcn_wmma_f32_16x16x4_f32(
                false, a, false, b, (short)0, acc[t], false, false);
        }
    }

    // Epilogue: store h, and fused dot products with att_src/att_dst.
    // C/D layout: VGPR r, lanes 0-15 -> M=r; lanes 16-31 -> M=r+8; N = lane%16 group.
    #pragma unroll
    for (int r = 0; r < 8; ++r) {
        const int orow = rowBase + r + khalf * 8;
        float ps = 0.f, pd = 0.f;
        #pragma unroll
        for (int t = 0; t < 8; ++t) {
            float v = acc[t][r];
            h[(size_t)orow * C + t * 16 + m16] = v;
            ps += v * attS[t];
            pd += v * attD[t];
        }
        // Reduce across the 16 lanes of each half-wave (xor masks stay in-group).
        #pragma unroll
        for (int off = 1; off < 16; off <<= 1) {
            ps += __shfl_xor(ps, off, 32);
            pd += __shfl_xor(pd, off, 32);
        }
        if (m16 == 0) {
            a_src[orow] = ps;
            a_dst[orow] = pd;
        }
    }
}

// ---------------------------------------------------------------------------
// Kernel 2: per-target 3-edge softmax + weighted gather of h rows.
// Every target i has exactly 3 in-edges: edge_index[0][i], edge_index[1][i], i.
// One wave per target; lanes handle 4 channels each (b128 accesses).
// ---------------------------------------------------------------------------
__global__ __launch_bounds__(256) void gat_edge_kernel(
    const float* __restrict__ h, const float* __restrict__ a_src,
    const float* __restrict__ a_dst, const int* __restrict__ edge_index,
    const float* __restrict__ bias, float* __restrict__ out)
{
    const int lane = threadIdx.x & 31;
    const int wave = threadIdx.x >> 5;
    const int i = blockIdx.x * 8 + wave;            // target row

    const int s0 = edge_index[i];                   // edge_index[0][i]
    const int s1 = edge_index[NE + i];              // edge_index[1][i]

    const float adi = a_dst[i];
    float e0 = a_src[s0] + adi;
    float e1 = a_src[s1] + adi;
    float e2 = a_src[i]  + adi;
    e0 = (e0 > 0.f) ? e0 : NEG_SLOPE * e0;
    e1 = (e1 > 0.f) ? e1 : NEG_SLOPE * e1;
    e2 = (e2 > 0.f) ? e2 : NEG_SLOPE * e2;

    const float m  = fmaxf(e0, fmaxf(e1, e2));
    const float x0 = expf(e0 - m);
    const float x1 = expf(e1 - m);
    const float x2 = expf(e2 - m);
    const float inv = 1.f / (x0 + x1 + x2);
    const float al0 = x0 * inv, al1 = x1 * inv, al2 = x2 * inv;

    const v4f* h0 = (const v4f*)(h + (size_t)s0 * C);
    const v4f* h1 = (const v4f*)(h + (size_t)s1 * C);
    const v4f* h2 = (const v4f*)(h + (size_t)i  * C);
    const v4f* b4 = (const v4f*)bias;

    v4f o = h0[lane] * al0 + h1[lane] * al1 + h2[lane] * al2 + b4[lane];
    ((v4f*)(out + (size_t)i * C))[lane] = o;
}

// ---------------------------------------------------------------------------
extern "C" void kernel_launch(void* const* d_in, const int* in_sizes, int n_in,
                              void* d_out, int out_size, void* d_ws, size_t ws_size,
                              hipStream_t stream) {
    const float* x        = (const float*)d_in[0];
    const int*   edge_idx = (const int*)  d_in[1];
    const float* W        = (const float*)d_in[2];
    const float* att_src  = (const float*)d_in[3];
    const float* att_dst  = (const float*)d_in[4];
    const float* bias     = (const float*)d_in[5];
    float* out = (float*)d_out;

    float* h      = (float*)d_ws;                   // [NE * C]
    float* a_srcW = h + (size_t)NE * C;             // [NE]
    float* a_dstW = a_srcW + NE;                    // [NE]

    gat_gemm_kernel<<<NE / 128, 256, 0, stream>>>(x, W, att_src, att_dst,
                                                  h, a_srcW, a_dstW);
    gat_edge_kernel<<<NE / 8, 256, 0, stream>>>(h, a_srcW, a_dstW,
                                                edge_idx, bias, out);
}